// Boundary_80676665688337
// MI455X (gfx1250) — compile-verified
//
#include <hip/hip_runtime.h>
#include <hip/hip_bf16.h>

#define EDT_LARGE 1e10f
#define FLT_BIG   3.402823466e38f

typedef __attribute__((ext_vector_type(2))) float v2f;
typedef __attribute__((ext_vector_type(8))) float v8f;

static constexpr int BATCH = 4, DX = 64, DY = 64, DZ = 64;
static constexpr int VOL = DX * DY * DZ;   // 262144
static constexpr int NTOT = BATCH * VOL;   // 1048576
static constexpr int NLINES = NTOT / 64;   // 16384
static constexpr int NTILES = NLINES / 64; // 256 (64 lines per block)
static constexpr int RED_BLOCKS = 1024;

// gfx1250 async global->LDS copy (ASYNCcnt-tracked); guarded so the file
// still compiles on toolchains that don't declare the builtins.
#if defined(__has_builtin)
#if __has_builtin(__builtin_amdgcn_global_load_async_to_lds_b32) && \
    __has_builtin(__builtin_amdgcn_s_wait_asynccnt)
#define USE_ASYNC_LDS 1
#endif
#endif

typedef __attribute__((address_space(1))) int glb_i32;
typedef __attribute__((address_space(3))) int lds_i32;

// ---------------------------------------------------------------------------
// 1) Initialize squared-distance fields: 0 at source voxels, LARGE elsewhere.
// ---------------------------------------------------------------------------
__global__ void init_edt_kernel(const int* __restrict__ target,
                                float* __restrict__ dpos,
                                float* __restrict__ dneg) {
    int i = blockIdx.x * blockDim.x + threadIdx.x;   // grid sized exactly
    int t = target[i];
    dpos[i] = (t == 0) ? 0.0f : EDT_LARGE;
    dneg[i] = (t == 0) ? EDT_LARGE : 0.0f;
}

// ---------------------------------------------------------------------------
// 2) Min-plus pass along one axis: d[i] = min_j d[j] + (i-j)^2 per 64-line.
//    Each block owns a 64-lines x 64-elements tile (disjoint across blocks;
//    all global reads complete before the barrier -> in-place safe).
//    Memory-linear tile fill (coalesced 128B/wave), LDS row stride 65 so both
//    compute access patterns are bank-conflict-free.
// ---------------------------------------------------------------------------
__global__ void minplus_kernel(float* __restrict__ d, int stride) {
    __shared__ float lds[64 * 65];
    const int tid = threadIdx.x;

    int base;
    if (stride == 1) {
        base = blockIdx.x * 4096;                 // 64 consecutive lines
    } else {
        const int chunks = stride >> 6;           // lo-chunks of 64
        const int hi  = blockIdx.x / chunks;
        const int loc = blockIdx.x - hi * chunks;
        base = hi * stride * 64 + loc * 64;
    }

    // Tile fill in memory-linear order: element k -> lds[(k>>6)*65 + (k&63)].
    for (int k = tid; k < 4096; k += 256) {
        const int gofs = (stride == 1) ? k : ((k >> 6) * stride + (k & 63));
        const int lidx = (k >> 6) * 65 + (k & 63);
#if USE_ASYNC_LDS
        __builtin_amdgcn_global_load_async_to_lds_b32(
            (glb_i32*)(d + base + gofs),
            (lds_i32*)&lds[lidx], 0, 0);
#else
        lds[lidx] = d[base + gofs];
#endif
    }
#if USE_ASYNC_LDS
    __builtin_amdgcn_s_wait_asynccnt(0);
#endif
    __syncthreads();

    if (stride == 1) {
        // lds[l*65 + i]; lanes parallel over i -> coalesced writes,
        // LDS reads are same-address wave broadcasts.
        const int i = tid & 63;
        for (int q = 0; q < 16; ++q) {
            const int l = (tid >> 6) + (q << 2);
            float best = FLT_BIG;
            for (int j = 0; j < 64; ++j) {
                const float diff = (float)(i - j);
                best = fminf(best, fmaf(diff, diff, lds[l * 65 + j]));
            }
            d[base + l * 64 + i] = best;
        }
    } else {
        // lds[i*65 + l]; thread owns column l: hoist each lds[j] over the
        // 16 outputs -> 64 ds_loads + 1024 fma/min per thread.
        const int l  = tid & 63;
        const int i0 = tid >> 6;                  // 0..3
        float best[16];
        for (int q = 0; q < 16; ++q) best[q] = FLT_BIG;
        for (int j = 0; j < 64; ++j) {
            const float dj = lds[j * 65 + l];
            for (int q = 0; q < 16; ++q) {
                const float diff = (float)(i0 + (q << 2) - j);
                best[q] = fminf(best[q], fmaf(diff, diff, dj));
            }
        }
        for (int q = 0; q < 16; ++q)              // coalesced: lanes vary l
            d[base + (i0 + (q << 2)) * stride + l] = best[q];
    }
}

// ---------------------------------------------------------------------------
// 3) sdf = sqrt(negdis)-sqrt(posdis), zeroed on inner boundary (branchless
//    clamped neighbor loads); multiply by output[:,1]; reduce.  Wave-level
//    reduction via V_WMMA_F32_16X16X4_F32: A(16x4) = 32 lane accumulators and 32 zeros,
//    B = ones => D[m][n] = row sums; lanes 0-15 hold rows 0-7, 16-31 rows 8-15.
// ---------------------------------------------------------------------------
__global__ void finalize_reduce_kernel(const float* __restrict__ output,
                                       const int* __restrict__ target,
                                       const float* __restrict__ dpos,
                                       const float* __restrict__ dneg,
                                       float* __restrict__ partials) {
    float acc = 0.0f;
    const int tid = blockIdx.x * blockDim.x + threadIdx.x;
    const int gstride = gridDim.x * blockDim.x;
    for (int v = tid; v < NTOT; v += gstride) {
        const int b = v >> 18;
        const int r = v & (VOL - 1);
        const int x = r >> 12;
        const int y = (r >> 6) & 63;
        const int z = r & 63;
        const float o = output[(b * 2 + 1) * VOL + r];   // output[b,1,x,y,z]

        const int t   = target[v];
        const int nxm = target[v - ((x > 0)  ? 4096 : 0)];
        const int nxp = target[v + ((x < 63) ? 4096 : 0)];
        const int nym = target[v - ((y > 0)  ? 64   : 0)];
        const int nyp = target[v + ((y < 63) ? 64   : 0)];
        const int nzm = target[v - ((z > 0)  ? 1    : 0)];
        const int nzp = target[v + ((z < 63) ? 1    : 0)];
        const bool bg = ((x > 0)  & (nxm == 0)) | ((x < 63) & (nxp == 0)) |
                        ((y > 0)  & (nym == 0)) | ((y < 63) & (nyp == 0)) |
                        ((z > 0)  & (nzm == 0)) | ((z < 63) & (nzp == 0));

        float sdf = sqrtf(dneg[v]) - sqrtf(dpos[v]);
        sdf = ((t != 0) && bg) ? 0.0f : sdf;             // inner boundary -> 0
        acc += o * sdf;
    }

    // All threads converged, full 256-thread blocks -> EXEC all 1s for WMMA.
    v2f a;    a[0]    = acc;  a[1]    = 0.0f;
    v2f ones; ones[0] = 1.0f; ones[1] = 1.0f;
    v8f c = {};
    c = __builtin_amdgcn_wmma_f32_16x16x4_f32(
        /*neg_a=*/false, a, /*neg_b=*/false, ones,
        /*c_mod=*/(short)0, c, /*reuse_a=*/false, /*reuse_b=*/false);
    float rowsum = c[0] + c[1] + c[2] + c[3] + c[4] + c[5] + c[6] + c[7];
    float wave_total = __shfl(rowsum, 0, 32) + __shfl(rowsum, 16, 32);

    __shared__ float wbuf[8];
    const int lane = threadIdx.x & 31;
    const int wv   = threadIdx.x >> 5;
    if (lane == 0) wbuf[wv] = wave_total;
    __syncthreads();
    if (threadIdx.x == 0) {
        float s = 0.0f;
        for (int w = 0; w < 8; ++w) s += wbuf[w];   // fixed order: deterministic
        partials[blockIdx.x] = s;
    }
}

// ---------------------------------------------------------------------------
// 4) Deterministic final tree reduction of the 1024 block partials -> mean.
// ---------------------------------------------------------------------------
__global__ void final_reduce_kernel(const float* __restrict__ partials,
                                    float* __restrict__ out) {
    __shared__ float buf[256];
    float s = 0.0f;
    for (int i = threadIdx.x; i < RED_BLOCKS; i += 256) s += partials[i];
    buf[threadIdx.x] = s;
    __syncthreads();
    for (int off = 128; off > 0; off >>= 1) {
        if (threadIdx.x < off) buf[threadIdx.x] += buf[threadIdx.x + off];
        __syncthreads();
    }
    if (threadIdx.x == 0) out[0] = buf[0] * (1.0f / (float)NTOT);
}

extern "C" void kernel_launch(void* const* d_in, const int* in_sizes, int n_in,
                              void* d_out, int out_size, void* d_ws, size_t ws_size,
                              hipStream_t stream) {
    (void)in_sizes; (void)n_in; (void)out_size; (void)ws_size;
    const float* output = (const float*)d_in[0];  // [4, 2, 64, 64, 64] f32
    const int*   target = (const int*)d_in[1];    // [4, 64, 64, 64]    i32

    float* dpos     = (float*)d_ws;               // 1M floats (4 MB)
    float* dneg     = dpos + NTOT;                // 1M floats (4 MB)
    float* partials = dneg + NTOT;                // 1024 floats
    float* out      = (float*)d_out;              // single scalar

    init_edt_kernel<<<NTOT / 256, 256, 0, stream>>>(target, dpos, dneg);

    // Separable exact squared-EDT: axes X (stride 4096), Y (64), Z (1).
    const int strides[3] = {DY * DZ, DZ, 1};
    for (int p = 0; p < 3; ++p) {
        minplus_kernel<<<NTILES, 256, 0, stream>>>(dpos, strides[p]);
        minplus_kernel<<<NTILES, 256, 0, stream>>>(dneg, strides[p]);
    }

    finalize_reduce_kernel<<<RED_BLOCKS, 256, 0, stream>>>(output, target, dpos,
                                                           dneg, partials);
    final_reduce_kernel<<<1, 256, 0, stream>>>(partials, out);
}